// clsAttn_Linear_36670430773948
// MI455X (gfx1250) — compile-verified
//
#include <hip/hip_runtime.h>
#include <math.h>

typedef __attribute__((ext_vector_type(2))) float v2f;
typedef __attribute__((ext_vector_type(8))) float v8f;

#define N_S    32
#define C_DIM  256
#define CLS_D  512
#define HW     4096
#define CHW    (C_DIM * HW)
#define EPS    1e-5f

// workspace layout (float offsets)
#define WS_K     0        // [32,256]
#define WS_Q     8192     // [32,256]
#define WS_Q2    16384    // [32,256]
#define WS_SUM1  24576    // [32,64]
#define WS_SSQ1  26624    // [32,64]
#define WS_ALPHA 28672    // [32,256]
#define WS_BETA  36864    // [32]
#define WS_SSUM  36896    // [32]
#define WS_SSQ2  36928    // [32]
#define WS_A2    36960    // [32,256]
#define WS_B2    45152    // [32,256]
#define WS_S     53344    // [32,4096]  (16B aligned: 53344 % 4 == 0)

// ---------------- init: zero the s-moment accumulators ----------------
__global__ void zero_kernel(float* __restrict__ ws) {
    int t = threadIdx.x;
    if (t < 64) ws[WS_SSUM + t] = 0.0f;   // covers WS_SSUM[32] + WS_SSQ2[32]
}

// ---------------- WMMA GEMM #1: K,Q embeddings ----------------
// z==0: K[n,c] = elu(cls[n]·wq[c] + bq[c]) + 1
// z==1: Q[n,c] = (elu(cls[n]·wk[c] + bk[c]) + 1) / 16
__global__ void emb_wmma_kernel(const float* __restrict__ cls,
                                const float* __restrict__ wq, const float* __restrict__ bq,
                                const float* __restrict__ wk, const float* __restrict__ bk,
                                float* __restrict__ ws) {
    const int lane  = threadIdx.x;            // wave32, all lanes active
    const int z     = blockIdx.z;
    const int nbase = blockIdx.x * 16;        // channel tile
    const int mbase = blockIdx.y * 16;        // sample tile
    const float* W    = z ? wk : wq;
    const float* bias = z ? bk : bq;
    float* dst = ws + (z ? WS_Q : WS_K);

    const int m  = lane & 15;                 // A row (sample), B col (channel)
    const int kk = (lane >> 4) * 2;           // K sub-offset per ISA f32 layout
    const float* arow = cls + (size_t)(mbase + m) * CLS_D + kk;
    const float* brow = W   + (size_t)(nbase + m) * CLS_D + kk;

    v8f acc = {0.f, 0.f, 0.f, 0.f, 0.f, 0.f, 0.f, 0.f};
    for (int k = 0; k < CLS_D; k += 4) {
        v2f a = *(const v2f*)(arow + k);
        v2f b = *(const v2f*)(brow + k);
        acc = __builtin_amdgcn_wmma_f32_16x16x4_f32(false, a, false, b,
                                                    (short)0, acc, false, false);
    }
    const int col   = lane & 15;
    const int rbase = (lane >> 4) * 8;
    const float bv = bias[nbase + col];
#pragma unroll
    for (int r = 0; r < 8; ++r) {
        const int row = rbase + r;
        float x = acc[r] + bv;
        float e = (x > 0.f) ? (x + 1.f) : expf(x);   // elu(x)+1
        if (z) e *= 0.0625f;                         // 1/sqrt(256)
        dst[(mbase + row) * C_DIM + (nbase + col)] = e;
    }
}

// ---------------- WMMA GEMM #2: q2 = Q @ conv_w^T ----------------
__global__ void q2_wmma_kernel(const float* __restrict__ conv_w, float* __restrict__ ws) {
    const int lane  = threadIdx.x;
    const int nbase = blockIdx.x * 16;        // output channel o tile
    const int mbase = blockIdx.y * 16;        // sample tile
    const float* Q = ws + WS_Q;
    float* q2      = ws + WS_Q2;

    const int m  = lane & 15;
    const int kk = (lane >> 4) * 2;
    const float* arow = Q      + (size_t)(mbase + m) * C_DIM + kk;
    const float* brow = conv_w + (size_t)(nbase + m) * C_DIM + kk;

    v8f acc = {0.f, 0.f, 0.f, 0.f, 0.f, 0.f, 0.f, 0.f};
    for (int k = 0; k < C_DIM; k += 4) {
        v2f a = *(const v2f*)(arow + k);
        v2f b = *(const v2f*)(brow + k);
        acc = __builtin_amdgcn_wmma_f32_16x16x4_f32(false, a, false, b,
                                                    (short)0, acc, false, false);
    }
    const int col   = lane & 15;
    const int rbase = (lane >> 4) * 8;
#pragma unroll
    for (int r = 0; r < 8; ++r)
        q2[(mbase + rbase + r) * C_DIM + (nbase + col)] = acc[r];
}

// ---------------- pass 1: per-(n,group) sum / sumsq of X ----------------
__global__ void xstats_kernel(const float* __restrict__ X, float* __restrict__ ws) {
    __shared__ float s1[256];
    __shared__ float s2[256];
    const int b = blockIdx.x;                 // n*64 + g
    const int t = threadIdx.x;
    const float4* p = (const float4*)(X + (size_t)b * (4 * HW));
    float sum = 0.f, ssq = 0.f;
#pragma unroll
    for (int i = 0; i < 16; ++i) {
        float4 v = p[t + i * 256];
        sum += v.x + v.y + v.z + v.w;
        ssq += v.x * v.x + v.y * v.y + v.z * v.z + v.w * v.w;
    }
    s1[t] = sum; s2[t] = ssq;
    __syncthreads();
    for (int st = 128; st > 0; st >>= 1) {
        if (t < st) { s1[t] += s1[t + st]; s2[t] += s2[t + st]; }
        __syncthreads();
    }
    if (t == 0) { ws[WS_SUM1 + b] = s1[0]; ws[WS_SSQ1 + b] = s2[0]; }
}

// ---------------- pass 2: fold GN1 into alpha[n,c], beta[n] ----------------
__global__ void alphabeta_kernel(const float* __restrict__ ln_w,
                                 const float* __restrict__ ln_b,
                                 float* __restrict__ ws) {
    __shared__ float red[256];
    const int n = blockIdx.x, c = threadIdx.x;
    const int g = c >> 2;
    const float inv = 1.0f / 16384.0f;
    float mu  = ws[WS_SUM1 + n * 64 + g] * inv;
    float var = ws[WS_SSQ1 + n * 64 + g] * inv - mu * mu;
    float rs  = rsqrtf(var + EPS);
    float kv  = ws[WS_K + n * C_DIM + c];
    float w   = ln_w[c];
    ws[WS_ALPHA + n * C_DIM + c] = kv * w * rs;
    red[c] = kv * (ln_b[c] - w * mu * rs);
    __syncthreads();
    for (int st = 128; st > 0; st >>= 1) {
        if (c < st) red[c] += red[c + st];
        __syncthreads();
    }
    if (c == 0) ws[WS_BETA + n] = red[0];
}

// ---------------- pass 3: s[n,hw] + HW moments of s ----------------
__global__ void s_kernel(const float* __restrict__ X, float* __restrict__ ws) {
    __shared__ float al[256];
    __shared__ float r1[256];
    __shared__ float r2[256];
    const int n = blockIdx.y;
    const int t = threadIdx.x;
    const int p = blockIdx.x * 256 + t;
    al[t] = ws[WS_ALPHA + n * C_DIM + t];
    __syncthreads();
    const float* xp = X + (size_t)n * CHW + p;
    float a0 = 0.f, a1 = 0.f, a2 = 0.f, a3 = 0.f;
    for (int c = 0; c < C_DIM; c += 4) {
        a0 += al[c + 0] * xp[(size_t)(c + 0) * HW];
        a1 += al[c + 1] * xp[(size_t)(c + 1) * HW];
        a2 += al[c + 2] * xp[(size_t)(c + 2) * HW];
        a3 += al[c + 3] * xp[(size_t)(c + 3) * HW];
    }
    float sv = a0 + a1 + a2 + a3 + ws[WS_BETA + n];
    ws[WS_S + n * HW + p] = sv;
    r1[t] = sv; r2[t] = sv * sv;
    __syncthreads();
    for (int st = 128; st > 0; st >>= 1) {
        if (t < st) { r1[t] += r1[t + st]; r2[t] += r2[t + st]; }
        __syncthreads();
    }
    if (t == 0) {
        atomicAdd(ws + WS_SSUM + n, r1[0]);
        atomicAdd(ws + WS_SSQ2 + n, r2[0]);
    }
}

// ---------------- pass 4: analytic GN2 -> A2[n,c], B2[n,c] ----------------
__global__ void coeff_kernel(const float* __restrict__ conv_b,
                             const float* __restrict__ gn_w,
                             const float* __restrict__ gn_b,
                             float* __restrict__ ws) {
    __shared__ float sa[256];
    __shared__ float sb[256];
    const int n = blockIdx.x, c = threadIdx.x;
    float a = ws[WS_Q2 + n * C_DIM + c];
    float b = conv_b[c];
    sa[c] = a; sb[c] = b;
    __syncthreads();
    const float m1 = ws[WS_SSUM + n] * (1.0f / 4096.0f);
    const float m2 = ws[WS_SSQ2 + n] * (1.0f / 4096.0f);
    const int base = (c >> 2) << 2;
    float mean = 0.f, ey2 = 0.f;
#pragma unroll
    for (int j = 0; j < 4; ++j) {
        float aj = sa[base + j], bj = sb[base + j];
        mean += aj * m1 + bj;
        ey2  += aj * aj * m2 + 2.0f * aj * bj * m1 + bj * bj;
    }
    mean *= 0.25f; ey2 *= 0.25f;
    float rs = rsqrtf(ey2 - mean * mean + EPS);
    ws[WS_A2 + n * C_DIM + c] = gn_w[c] * a * rs;
    ws[WS_B2 + n * C_DIM + c] = gn_w[c] * (b - mean) * rs + gn_b[c];
}

// ---------------- pass 5: out = A2*s + B2 + X ----------------
__global__ void out_kernel(const float* __restrict__ X, const float* __restrict__ ws,
                           float* __restrict__ out) {
    const int bid = blockIdx.x;
    const int quarter = bid & 3;
    const int nc = bid >> 2;          // n*256 + c
    const int n  = nc >> 8;
    const int off = quarter * 1024 + threadIdx.x * 4;
    const float A = ws[WS_A2 + nc];
    const float B = ws[WS_B2 + nc];
    const size_t xi = (size_t)nc * HW + off;
    float4 xv = *(const float4*)(X + xi);
    float4 sv = *(const float4*)(ws + WS_S + n * HW + off);
    float4 o;
    o.x = A * sv.x + B + xv.x;
    o.y = A * sv.y + B + xv.y;
    o.z = A * sv.z + B + xv.z;
    o.w = A * sv.w + B + xv.w;
    *(float4*)(out + xi) = o;
}

extern "C" void kernel_launch(void* const* d_in, const int* in_sizes, int n_in,
                              void* d_out, int out_size, void* d_ws, size_t ws_size,
                              hipStream_t stream) {
    const float* X      = (const float*)d_in[0];
    const float* cls    = (const float*)d_in[1];
    const float* ln_w   = (const float*)d_in[2];
    const float* ln_b   = (const float*)d_in[3];
    const float* wq     = (const float*)d_in[4];
    const float* bq     = (const float*)d_in[5];
    const float* wk     = (const float*)d_in[6];
    const float* bk     = (const float*)d_in[7];
    const float* conv_w = (const float*)d_in[8];
    const float* conv_b = (const float*)d_in[9];
    const float* gn_w   = (const float*)d_in[10];
    const float* gn_b   = (const float*)d_in[11];
    float* ws  = (float*)d_ws;
    float* out = (float*)d_out;

    zero_kernel<<<1, 64, 0, stream>>>(ws);
    emb_wmma_kernel<<<dim3(16, 2, 2), 32, 0, stream>>>(cls, wq, bq, wk, bk, ws);
    q2_wmma_kernel<<<dim3(16, 2), 32, 0, stream>>>(conv_w, ws);
    xstats_kernel<<<2048, 256, 0, stream>>>(X, ws);
    alphabeta_kernel<<<32, 256, 0, stream>>>(ln_w, ln_b, ws);
    s_kernel<<<dim3(16, 32), 256, 0, stream>>>(X, ws);
    coeff_kernel<<<32, 256, 0, stream>>>(conv_b, gn_w, gn_b, ws);
    out_kernel<<<32768, 256, 0, stream>>>(X, ws, out);
}